// GATModel_28114855919705
// MI455X (gfx1250) — compile-verified
//
#include <hip/hip_runtime.h>
#include <math.h>

// ---------------- problem constants (from reference) ----------------
constexpr int G     = 512;
constexpr int P     = 128;
constexpr int NA    = 5;          // agents per graph
constexpr int EPER  = 1024;
constexpr int N     = G * P;      // 65536 nodes
constexpr int E     = G * EPER;   // 524288 edges
constexpr int H     = 8;          // heads
constexpr int C     = 16;         // hid per head
constexpr int F     = H * C;      // 128 channels
constexpr int E2    = E + N;      // edges + self loops
constexpr int INCH  = 4;
constexpr int OUTCH = 2;
constexpr int MHEAD = G * NA;     // 2560 rows into MLP head
constexpr int K1    = 2 * F;      // 256 (fc1 in)
constexpr int F1    = 64;         // fc1 out
constexpr float NEG_SLOPE = 0.2f;

constexpr int TPB = 256;
static inline unsigned gridFor(long long n) { return (unsigned)((n + TPB - 1) / TPB); }

typedef float v2f __attribute__((ext_vector_type(2)));
typedef float v8f __attribute__((ext_vector_type(8)));

// ---------------- helpers ----------------
__device__ __forceinline__ unsigned ordEnc(float f) {
    unsigned u = __float_as_uint(f);
    return (u & 0x80000000u) ? ~u : (u | 0x80000000u);
}
__device__ __forceinline__ float ordDec(unsigned u) {
    unsigned b = (u & 0x80000000u) ? (u & 0x7FFFFFFFu) : ~u;
    return __uint_as_float(b);
}
// edge e -> (src, dst, edge_attr); e >= E are self loops (layers 2/3 only)
__device__ __forceinline__ void edgeSD(int e, const int* __restrict__ src,
                                       const int* __restrict__ dst,
                                       const float* __restrict__ ea,
                                       const float* __restrict__ loopea,
                                       int& s, int& d, float& av) {
    if (e < E) { s = src[e]; d = dst[e]; av = ea[e]; }
    else       { int n = e - E; s = n; d = n; av = loopea[n]; }
}

// ---------------- WMMA fp32 GEMM: C[M,Fout] = A[M,K] @ W[K,Fout] ----------------
// One wave computes one 16x16 output tile via V_WMMA_F32_16X16X4_F32 (K step 4).
// A fragment (ISA 7.12.2, 32-bit A 16x4): lanes 0-15 -> K=k0,k0+1 ; lanes 16-31 -> K=k0+2,k0+3.
// B/C/D row-striped across lanes. Optional fused bias + ReLU at store.
__global__ void gemm_wmma_f32(const float* __restrict__ Amat, const float* __restrict__ Wmat,
                              float* __restrict__ Cmat, const float* __restrict__ bias,
                              int M, int K, int Fout, int doRelu) {
    const int wavesPerBlock = blockDim.x >> 5;
    const int tile = blockIdx.x * wavesPerBlock + ((int)threadIdx.x >> 5);
    const int tilesN = Fout >> 4;
    const int totalTiles = (M >> 4) * tilesN;
    if (tile >= totalTiles) return;          // wave-uniform exit (EXEC stays all-1 for WMMA)
    const int m0 = (tile / tilesN) << 4;
    const int n0 = (tile % tilesN) << 4;
    const int lane = (int)threadIdx.x & 31;
    const int half = lane >> 4;              // 0: K lo pair, 1: K hi pair
    const int lr   = lane & 15;
    v8f acc = {};
    const float* __restrict__ arow = Amat + (size_t)(m0 + lr) * K;
    const float* __restrict__ bcol = Wmat + n0 + lr;
    for (int k0 = 0; k0 < K; k0 += 4) {
        const int ka = k0 + (half << 1);
        v2f a, b;
        a[0] = arow[ka];
        a[1] = arow[ka + 1];
        b[0] = bcol[(size_t)ka * Fout];
        b[1] = bcol[(size_t)(ka + 1) * Fout];
        acc = __builtin_amdgcn_wmma_f32_16x16x4_f32(
            /*neg_a=*/false, a, /*neg_b=*/false, b,
            /*c_mod=*/(short)0, acc, /*reuse_a=*/false, /*reuse_b=*/false);
    }
#pragma unroll
    for (int v = 0; v < 8; ++v) {
        const int row = m0 + v + (half << 3);
        const int col = n0 + lr;
        float val = acc[v];
        if (bias)   val += bias[col];
        if (doRelu) val  = fmaxf(val, 0.f);
        Cmat[(size_t)row * Fout + col] = val;
    }
}

// ---------------- attention scalar kernels ----------------
// al_src[n,h] = <h[n,h,:], att_src[h,:]> ; al_dst likewise
__global__ void node_alpha_k(const float* __restrict__ ht, const float* __restrict__ att_src,
                             const float* __restrict__ att_dst,
                             float* __restrict__ al_src, float* __restrict__ al_dst) {
    int i = blockIdx.x * blockDim.x + threadIdx.x;
    if (i >= N * H) return;
    int n = i / H, h = i % H;
    const float* hp = ht + (size_t)n * F + h * C;
    float s = 0.f, d = 0.f;
#pragma unroll
    for (int c = 0; c < C; ++c) {
        float v = hp[c];
        s += v * att_src[h * C + c];
        d += v * att_dst[h * C + c];
    }
    al_src[i] = s;
    al_dst[i] = d;
}

// we_dot[h] = <We[h*C..], att_edge[h,:]>  (edge_attr is scalar so al_e = ea * we_dot[h])
__global__ void wedot_k(const float* __restrict__ We, const float* __restrict__ att_edge,
                        float* __restrict__ wedot) {
    int h = threadIdx.x;
    if (h < H) {
        float s = 0.f;
#pragma unroll
        for (int c = 0; c < C; ++c) s += We[h * C + c] * att_edge[h * C + c];
        wedot[h] = s;
    }
}

__global__ void init_amax_k(unsigned* __restrict__ amax) {
    int i = blockIdx.x * blockDim.x + threadIdx.x;
    if (i < N * H) amax[i] = ordEnc(-INFINITY);
}

// pass 1: alpha = leaky_relu(al_src[s]+al_dst[d]+ea*wedot) ; segment max via ordered-u32 atomicMax
__global__ void edge_alpha_k(const int* __restrict__ src, const int* __restrict__ dst,
                             const float* __restrict__ ea, const float* __restrict__ loopea,
                             const float* __restrict__ al_src, const float* __restrict__ al_dst,
                             const float* __restrict__ wedot,
                             float* __restrict__ alpha, unsigned* __restrict__ amax, int nE) {
    int i = blockIdx.x * blockDim.x + threadIdx.x;
    if (i >= nE * H) return;
    int e = i / H, h = i % H;
    int s, d; float av;
    edgeSD(e, src, dst, ea, loopea, s, d, av);
    float a = al_src[s * H + h] + al_dst[d * H + h] + av * wedot[h];
    a = (a > 0.f) ? a : NEG_SLOPE * a;
    alpha[i] = a;
    atomicMax(&amax[d * H + h], ordEnc(a));
}

// pass 2: ex = exp(alpha - amax[d]) ; segment sum of ex into denom
__global__ void edge_exp_k(const int* __restrict__ src, const int* __restrict__ dst,
                           const float* __restrict__ ea, const float* __restrict__ loopea,
                           float* __restrict__ alpha, const unsigned* __restrict__ amax,
                           float* __restrict__ denom, int nE) {
    int i = blockIdx.x * blockDim.x + threadIdx.x;
    if (i >= nE * H) return;
    int e = i / H, h = i % H;
    int s, d; float av;
    edgeSD(e, src, dst, ea, loopea, s, d, av);
    float ex = __expf(alpha[i] - ordDec(amax[d * H + h]));
    alpha[i] = ex;
    atomicAdd(&denom[d * H + h], ex);
}

// pass 3: out[d,f] += h[s,f] * ex/(denom[d,h]+eps)   (streaming, atomic-add bound)
__global__ void edge_agg_k(const int* __restrict__ src, const int* __restrict__ dst,
                           const float* __restrict__ ea, const float* __restrict__ loopea,
                           const float* __restrict__ ht, const float* __restrict__ alpha,
                           const float* __restrict__ denom, float* __restrict__ hout, int nE) {
    int i = blockIdx.x * blockDim.x + threadIdx.x;
    if (i >= nE * F) return;
    int e = i / F, f = i % F, h = f / C;
    int s, d; float av;
    edgeSD(e, src, dst, ea, loopea, s, d, av);
    float att = alpha[e * H + h] / (denom[d * H + h] + 1e-16f);
    atomicAdd(&hout[(size_t)d * F + f], ht[(size_t)s * F + f] * att);
}

__global__ void finalize_k(float* __restrict__ hout, const float* __restrict__ bias) {
    int i = blockIdx.x * blockDim.x + threadIdx.x;
    if (i >= N * F) return;
    float v = hout[i] + bias[i % F];
    hout[i] = fmaxf(v, 0.f);
}

// ---------------- self-loop edge_attr = mean of incoming ----------------
__global__ void loop_stats_k(const int* __restrict__ dst, const float* __restrict__ ea,
                             float* __restrict__ easum, float* __restrict__ cnt) {
    int e = blockIdx.x * blockDim.x + threadIdx.x;
    if (e >= E) return;
    int d = dst[e];
    atomicAdd(&easum[d], ea[e]);
    atomicAdd(&cnt[d], 1.f);
}
__global__ void loop_final_k(float* __restrict__ loopea, const float* __restrict__ easum,
                             const float* __restrict__ cnt) {
    int n = blockIdx.x * blockDim.x + threadIdx.x;
    if (n < N) loopea[n] = easum[n] / fmaxf(cnt[n], 1.f);
}

// ---------------- head ----------------
__global__ void pool_k(const float* __restrict__ hfin, float* __restrict__ gemb) {
    int i = blockIdx.x * blockDim.x + threadIdx.x;
    if (i >= G * F) return;
    int g = i / F, f = i % F;
    float s = 0.f;
    const float* base = hfin + (size_t)g * P * F + f;
    for (int p = 0; p < P; ++p) s += base[(size_t)p * F];
    gemb[i] = s * (1.f / (float)P);
}
__global__ void comb_k(const float* __restrict__ hfin, const float* __restrict__ gemb,
                       float* __restrict__ comb) {
    int i = blockIdx.x * blockDim.x + threadIdx.x;
    if (i >= MHEAD * K1) return;
    int m = i / K1, k = i % K1;
    int g = m / NA, a = m % NA;
    comb[i] = (k < F) ? hfin[(size_t)(g * P + a) * F + k] : gemb[g * F + (k - F)];
}
__global__ void fc2_k(const float* __restrict__ hmid, const float* __restrict__ w,
                      const float* __restrict__ b, float* __restrict__ out) {
    int i = blockIdx.x * blockDim.x + threadIdx.x;
    if (i >= MHEAD * OUTCH) return;
    int m = i / OUTCH, o = i % OUTCH;
    float s = b[o];
    const float* hp = hmid + (size_t)m * F1;
#pragma unroll 8
    for (int k = 0; k < F1; ++k) s += hp[k] * w[k * OUTCH + o];
    out[i] = s;
}

// ---------------- launch ----------------
extern "C" void kernel_launch(void* const* d_in, const int* in_sizes, int n_in,
                              void* d_out, int out_size, void* d_ws, size_t ws_size,
                              hipStream_t stream) {
    const float* x    = (const float*)d_in[0];
    const int*   eidx = (const int*)  d_in[1];
    const int*   src  = eidx;
    const int*   dst  = eidx + E;
    const float* ea   = (const float*)d_in[2];
    // params: c1(3..8) c2(9..14) c3(15..20), fc1_w/b(21,22), fc2_w/b(23,24)
    const float* cW [3] = {(const float*)d_in[3],  (const float*)d_in[9],  (const float*)d_in[15]};
    const float* cAS[3] = {(const float*)d_in[4],  (const float*)d_in[10], (const float*)d_in[16]};
    const float* cAD[3] = {(const float*)d_in[5],  (const float*)d_in[11], (const float*)d_in[17]};
    const float* cWe[3] = {(const float*)d_in[6],  (const float*)d_in[12], (const float*)d_in[18]};
    const float* cAE[3] = {(const float*)d_in[7],  (const float*)d_in[13], (const float*)d_in[19]};
    const float* cB [3] = {(const float*)d_in[8],  (const float*)d_in[14], (const float*)d_in[20]};
    const float* fc1w = (const float*)d_in[21];
    const float* fc1b = (const float*)d_in[22];
    const float* fc2w = (const float*)d_in[23];
    const float* fc2b = (const float*)d_in[24];
    float* out = (float*)d_out;

    // workspace carve-up (floats)
    float* ws = (float*)d_ws;
    size_t off = 0;
    auto alloc = [&](size_t n) { float* p = ws + off; off += (n + 7) & ~(size_t)7; return p; };
    float*    hT     = alloc((size_t)N * F);   // transformed features (pre-aggregation)
    float*    hA     = alloc((size_t)N * F);   // layer outputs ping
    float*    hB     = alloc((size_t)N * F);   // layer outputs pong
    float*    alsrc  = alloc((size_t)N * H);
    float*    aldst  = alloc((size_t)N * H);
    float*    alpha  = alloc((size_t)E2 * H);
    unsigned* amax   = (unsigned*)alloc((size_t)N * H);
    float*    denom  = alloc((size_t)N * H);
    float*    loopea = alloc(N);
    float*    easum  = alloc(N);
    float*    cntb   = alloc(N);
    float*    wedot  = alloc(8);
    float*    gemb   = alloc((size_t)G * F);
    float*    comb   = alloc((size_t)MHEAD * K1);
    float*    hmid   = alloc((size_t)MHEAD * F1);

    auto run_layer = [&](const float* hin, int Kin, int li, const float* loopPtr, int nE,
                         float* hout) {
        // 1) node transform via fp32 WMMA
        {
            int tiles = (N / 16) * (F / 16);
            gemm_wmma_f32<<<(tiles + 3) / 4, 128, 0, stream>>>(hin, cW[li], hT, nullptr,
                                                               N, Kin, F, 0);
        }
        // 2) per-node attention logits
        node_alpha_k<<<gridFor((long long)N * H), TPB, 0, stream>>>(hT, cAS[li], cAD[li],
                                                                    alsrc, aldst);
        wedot_k<<<1, 32, 0, stream>>>(cWe[li], cAE[li], wedot);
        // 3) segment softmax over incoming edges
        init_amax_k<<<gridFor((long long)N * H), TPB, 0, stream>>>(amax);
        hipMemsetAsync(denom, 0, (size_t)N * H * sizeof(float), stream);
        edge_alpha_k<<<gridFor((long long)nE * H), TPB, 0, stream>>>(
            src, dst, ea, loopPtr, alsrc, aldst, wedot, alpha, amax, nE);
        edge_exp_k<<<gridFor((long long)nE * H), TPB, 0, stream>>>(
            src, dst, ea, loopPtr, alpha, amax, denom, nE);
        // 4) weighted aggregation (dominant HBM/atomic cost)
        hipMemsetAsync(hout, 0, (size_t)N * F * sizeof(float), stream);
        edge_agg_k<<<gridFor((long long)nE * F), TPB, 0, stream>>>(
            src, dst, ea, loopPtr, hT, alpha, denom, hout, nE);
        // 5) bias + relu
        finalize_k<<<gridFor((long long)N * F), TPB, 0, stream>>>(hout, cB[li]);
    };

    // self-loop edge_attr (fill_value='mean'), computed once
    hipMemsetAsync(easum, 0, N * sizeof(float), stream);
    hipMemsetAsync(cntb, 0, N * sizeof(float), stream);
    loop_stats_k<<<gridFor(E), TPB, 0, stream>>>(dst, ea, easum, cntb);
    loop_final_k<<<gridFor(N), TPB, 0, stream>>>(loopea, easum, cntb);

    run_layer(x,  INCH, 0, nullptr, E,  hA);   // conv1: no self loops
    run_layer(hA, F,    1, loopea,  E2, hB);   // conv2
    run_layer(hB, F,    2, loopea,  E2, hA);   // conv3

    // head: mean-pool, concat(agent, graph), fc1 (WMMA + fused bias/relu), fc2
    pool_k<<<gridFor((long long)G * F), TPB, 0, stream>>>(hA, gemb);
    comb_k<<<gridFor((long long)MHEAD * K1), TPB, 0, stream>>>(hA, gemb, comb);
    {
        int tiles = (MHEAD / 16) * (F1 / 16);
        gemm_wmma_f32<<<(tiles + 3) / 4, 128, 0, stream>>>(comb, fc1w, hmid, fc1b,
                                                           MHEAD, K1, F1, 1);
    }
    fc2_k<<<gridFor((long long)MHEAD * OUTCH), TPB, 0, stream>>>(hmid, fc2w, fc2b, out);
}